// ECE_v2_14740327760392
// MI455X (gfx1250) — compile-verified
//
#include <hip/hip_runtime.h>
#include <stdint.h>

// ECE (expected calibration error) for [N=2e6, C=128] fp32 softmaxes.
// Memory-bound: 1.02 GB streamed once -> ~44us floor at 23.3 TB/s.
// CDNA5 path: TDM tensor_load_to_lds, 3-buffer / 2-ahead async pipeline
// (TENSORcnt) + wave32 shfl_xor butterfly max/argmax reduction.
// Fully deterministic: register bin accumulators (lane b owns bin b),
// fixed-order block reduce, fixed-order final reduce. No float atomics.

#define ROWS     32    // rows per tile (32 x 128 x 4B = 16 KB per buffer)
#define NBUF     3     // triple buffer, 2 tiles in flight
#define NWAVES   8     // 256 threads / wave32
#define NBINS    15
#define TILEB    (ROWS * 128 * 4)   // bytes per LDS tile buffer

#if defined(__AMDGCN__) && __has_builtin(__builtin_amdgcn_tensor_load_to_lds)
#define USE_TDM 1
#else
#define USE_TDM 0
#endif

#if defined(__AMDGCN__) && __has_builtin(__builtin_amdgcn_s_wait_tensorcnt)
#define WAIT_TENSORCNT(n) __builtin_amdgcn_s_wait_tensorcnt(n)
#elif defined(__AMDGCN__)
#define WAIT_TENSORCNT(n) asm volatile("s_wait_tensorcnt %0" :: "n"(n))
#else
#define WAIT_TENSORCNT(n)
#endif

#if USE_TDM
typedef unsigned int v4u __attribute__((ext_vector_type(4)));
typedef int          v4i __attribute__((ext_vector_type(4)));
typedef int          v8i __attribute__((ext_vector_type(8)));

// Pack a Tensor-DMA descriptor (D#) for a 2D tile: ROWS rows x 128 f32,
// row stride 128 elements, zero-fill past tensor_dim1 rows (ISA 08 §8.3-8.4).
// All fields are wave-uniform 32-bit values -> stays on the SALU.
__device__ __forceinline__ void tdm_issue_tile(const float* sm, long long rowStart,
                                               long long N, uint32_t ldsOff) {
    unsigned long long ga = (unsigned long long)(uintptr_t)(sm + rowStart * 128);
    uint32_t td1 = (uint32_t)(N - rowStart);                    // rows remaining (< 2^31)
    v4u g0;
    g0.x = 1u;                                                  // count=1 (valid), no gather
    g0.y = ldsOff;                                              // lds_addr (bytes)
    g0.z = (uint32_t)(ga & 0xFFFFFFFFull);                      // global_addr[31:0]
    g0.w = (uint32_t)((ga >> 32) & 0x01FFFFFFull) | (2u << 30); // addr[56:32] | type=2
    v8i g1;
    g1[0] = (int)(2u << 16);                                    // data_size=4B, wg_mask=0
    g1[1] = (int)(128u << 16);                                  // tensor_dim0 lo16 = 128
    g1[2] = (int)((td1 & 0xFFFFu) << 16);                       // dim0 hi=0 | tensor_dim1 lo16
    g1[3] = (int)((td1 >> 16) | (128u << 16));                  // dim1 hi16 | tile_dim0=128
    g1[4] = (int)(ROWS & 0xFFFF);                               // tile_dim1=ROWS, tile_dim2=0
    g1[5] = 128;                                                // tensor_dim0_stride = 128 elems
    g1[6] = 0;                                                  // stride hi | dim1_stride lo
    g1[7] = 0;
    v4i z4 = {0, 0, 0, 0};                                      // groups 2/3 unused (2D tensor)
#if __clang_major__ >= 23
    v8i z8 = {0, 0, 0, 0, 0, 0, 0, 0};
    __builtin_amdgcn_tensor_load_to_lds(g0, g1, z4, z4, z8, 0);
#else
    __builtin_amdgcn_tensor_load_to_lds(g0, g1, z4, z4, 0);
#endif
}
#endif // USE_TDM

// One row: lane holds cols 4*lane..4*lane+3; butterfly -> all lanes get
// (max, first argmax). Lane b accumulates bin b (deterministic, no atomics).
__device__ __forceinline__ void accum_row(float4 v, int lane, int label,
                                          float& c_sum, float& a_sum) {
    int base = lane * 4;
    float m = v.x; int mi = base;
    if (v.y > m) { m = v.y; mi = base + 1; }
    if (v.z > m) { m = v.z; mi = base + 2; }
    if (v.w > m) { m = v.w; mi = base + 3; }
#pragma unroll
    for (int off = 16; off > 0; off >>= 1) {
        float om = __shfl_xor(m, off, 32);
        int   oi = __shfl_xor(mi, off, 32);
        if (om > m || (om == m && oi < mi)) { m = om; mi = oi; } // first-occurrence argmax
    }
    int b = (int)ceilf(m * (float)NBINS) - 1;                   // bin: (i/15, (i+1)/15]
    b = b < 0 ? 0 : (b > NBINS - 1 ? NBINS - 1 : b);
    if (lane == b) {
        c_sum += m;
        a_sum += (mi == label) ? 1.0f : 0.0f;
    }
}

__global__ void __launch_bounds__(256)
ece_partial_kernel(const float* __restrict__ sm, const int* __restrict__ labels,
                   long long N, float* __restrict__ partials) {
    const int lane = threadIdx.x & 31;
    const int wid  = threadIdx.x >> 5;
    const long long numTiles = (N + ROWS - 1) / ROWS;
    float c_sum = 0.0f, a_sum = 0.0f;

#if USE_TDM
    __shared__ float tile[NBUF][ROWS][128];                     // 3 x 16 KB ring
    const uint32_t lds0 = (uint32_t)(uintptr_t)&tile[0][0][0];
    const long long G = (long long)gridDim.x;

    int cur = 0;                                                // buffer being consumed
    long long t = (long long)blockIdx.x;
    if (wid == 0) {                                             // prologue: 2 tiles in flight
        if (t < numTiles)     tdm_issue_tile(sm, t * ROWS, N, lds0);
        if (t + G < numTiles) tdm_issue_tile(sm, (t + G) * ROWS, N, lds0 + TILEB);
    }

    for (; t < numTiles; t += G) {
        const long long tn1 = t + G;
        const long long tn2 = t + 2 * G;
        if (wid == 0) {
            if (tn2 < numTiles) {
                int ib = cur + 2; if (ib >= NBUF) ib -= NBUF;   // issue 2 ahead
                tdm_issue_tile(sm, tn2 * ROWS, N, lds0 + (uint32_t)ib * TILEB);
                WAIT_TENSORCNT(2);                              // {t} done, {t+G,t+2G} in flight
            } else if (tn1 < numTiles) {
                WAIT_TENSORCNT(1);                              // {t} done, {t+G} in flight
            } else {
                WAIT_TENSORCNT(0);                              // all done
            }
        }
        __syncthreads();                                        // publish current buffer
        const long long rowBase = t * ROWS;
#pragma unroll
        for (int i = 0; i < ROWS / NWAVES; ++i) {
            const int r = wid + i * NWAVES;
            const long long row = rowBase + r;
            if (row < N) {
                const float4 v = *(const float4*)&tile[cur][r][lane * 4]; // ds_load_b128
                accum_row(v, lane, labels[row], c_sum, a_sum);
            }
        }
        __syncthreads();                                        // reads done before reuse
        cur = (cur + 1 == NBUF) ? 0 : cur + 1;
    }
#else
    for (long long t = (long long)blockIdx.x; t < numTiles; t += (long long)gridDim.x) {
        const long long rowBase = t * ROWS;
#pragma unroll
        for (int i = 0; i < ROWS / NWAVES; ++i) {
            const int r = wid + i * NWAVES;
            const long long row = rowBase + r;
            if (row < N) {
                const float4 v = *(const float4*)(sm + row * 128 + lane * 4); // global b128
                accum_row(v, lane, labels[row], c_sum, a_sum);
            }
        }
    }
#endif

    // Deterministic fixed-order block reduction over the 8 waves.
    __shared__ float s_c[NWAVES][16];
    __shared__ float s_a[NWAVES][16];
    if (lane < 16) { s_c[wid][lane] = c_sum; s_a[wid][lane] = a_sum; }
    __syncthreads();
    if (threadIdx.x < 16) {
        float c = 0.0f, a = 0.0f;
#pragma unroll
        for (int w = 0; w < NWAVES; ++w) { c += s_c[w][threadIdx.x]; a += s_a[w][threadIdx.x]; }
        partials[(long long)blockIdx.x * 32 + threadIdx.x]      = c;
        partials[(long long)blockIdx.x * 32 + 16 + threadIdx.x] = a;
    }
}

__global__ void __launch_bounds__(32)
ece_finalize_kernel(const float* __restrict__ partials, int nBlocks,
                    long long N, float* __restrict__ out) {
    __shared__ float diffs[NBINS];
    const int t = threadIdx.x;
    if (t < NBINS) {
        float c = 0.0f, a = 0.0f;
        for (int g = 0; g < nBlocks; ++g) {                     // fixed order -> deterministic
            c += partials[(long long)g * 32 + t];
            a += partials[(long long)g * 32 + 16 + t];
        }
        diffs[t] = fabsf(c - a);
    }
    __syncthreads();
    if (t == 0) {
        float s = 0.0f;
        for (int b = 0; b < NBINS; ++b) s += diffs[b];
        out[0] = s / (float)N;
    }
}

extern "C" void kernel_launch(void* const* d_in, const int* in_sizes, int n_in,
                              void* d_out, int out_size, void* d_ws, size_t ws_size,
                              hipStream_t stream) {
    const float* sm     = (const float*)d_in[0];
    const int*   labels = (const int*)d_in[1];   // integer input -> const int*
    const long long N   = (long long)in_sizes[1];
    float* out      = (float*)d_out;
    float* partials = (float*)d_ws;

    const long long numTiles = (N + ROWS - 1) / ROWS;
    long long g = 2048;                                   // enough blocks to cover all WGPs
    if (g > numTiles) g = numTiles;
    const long long wsCap = (long long)(ws_size / (32 * sizeof(float)));
    if (g > wsCap) g = wsCap;
    if (g < 1) g = 1;
    const int G = (int)g;

    ece_partial_kernel<<<G, 256, 0, stream>>>(sm, labels, N, partials);
    ece_finalize_kernel<<<1, 32, 0, stream>>>(partials, G, N, out);
}